// LaplacianLoss_35141422416050
// MI455X (gfx1250) — compile-verified
//
#include <hip/hip_runtime.h>

// LaplacianLoss on MI455X (gfx1250, wave32).
// loss = sum((dx - (A·dx)/deg)^2), dx = pred2-pred1, deg = column sums of A.
// Memory-bound: A = 512MB streamed once -> ~22us floor at 23.3 TB/s.
// Fused single pass: V_WMMA_F32_16X16X4_F32 for the row contraction,
// ds b128 column sums + ds_add_f32 for deg, and double-buffered
// GLOBAL_LOAD_ASYNC_TO_LDS_B128 staging of A tiles (ASYNCcnt-tracked),
// falling back to register-prefetch staging if the async builtins are absent.
//
// Workspace layout (floats): dx[8*4096*3] | neigh[8*4096*3] | degpart[8*64*4096]
// total ~9.2 MB.

#define NB 8
#define NN 4096
#define ND 3
#define BR 64              // rows per workgroup
#define TW 64              // columns per tile
#define ATW 68             // padded LDS row stride (dwords) -> conflict-free wmma reads
#define TPB 128            // 4 waves of 32
#define NTILE (NN / TW)    // 64 column tiles
#define WGPB (NN / BR)     // 64 workgroups per batch

typedef __attribute__((ext_vector_type(2))) float v2f;
typedef __attribute__((ext_vector_type(8))) float v8f;

#if __has_builtin(__builtin_amdgcn_global_load_async_to_lds_b128) && \
    __has_builtin(__builtin_amdgcn_s_wait_asynccnt)
#define USE_ASYNC 1
#else
#define USE_ASYNC 0
#endif

#if USE_ASYNC
// Builtin signature (from hipcc diagnostic): int4* in AS1 (global) / AS3 (LDS).
typedef int v4i __attribute__((__vector_size__(4 * sizeof(int))));
typedef __attribute__((address_space(1))) v4i* g4p;
typedef __attribute__((address_space(3))) v4i* l4p;
#define AS1(p) ((g4p)(uintptr_t)(p))
#define AS3(p) ((l4p)(uintptr_t)(uint32_t)(uintptr_t)(p))
#endif

__global__ void lap_prep(const float* __restrict__ p1, const float* __restrict__ p2,
                         float* __restrict__ dxw, float* __restrict__ out) {
  int gid = blockIdx.x * blockDim.x + threadIdx.x;
  if (gid < NB * NN * ND) dxw[gid] = p2[gid] - p1[gid];
  if (gid == 0) out[0] = 0.0f;
}

__global__ __launch_bounds__(TPB) void lap_main(const float* __restrict__ A,
                                                const float* __restrict__ dxw,
                                                float* __restrict__ neighw,
                                                float* __restrict__ degp) {
  __shared__ float At[2][BR * ATW];       // double-buffered 64x64 A tile, padded rows
  __shared__ float dxt[2][TW * ND + 4];   // dx tile; index TW*ND is a guaranteed-zero slot
  __shared__ float degl[NN];              // per-WG partial column sums

  const int t    = threadIdx.x;
  const int b    = blockIdx.x / WGPB;
  const int rb   = blockIdx.x % WGPB;
  const int r0   = rb * BR;

  const int lane = t & 31;
  const int wave = t >> 5;
  const int mi   = lane & 15;             // row-in-wave-tile (A) / output column n (B,C)
  const int kk   = (lane >> 4) << 1;      // 0 or 2 : K-pair per half-wave (ISA layout)
  const bool nv  = (mi < ND);
  const int aRowBase = (wave * 16 + mi) * ATW + kk;
  const int zslot = TW * ND;

  const int c4 = (t & 15) << 2;           // deg: 4 adjacent columns
  const int rg = (t >> 4) << 3;           // deg: 8-row group start

  for (int q = 0; q < NN / TPB; ++q) degl[q * TPB + t] = 0.0f;
  if (t == 0) { dxt[0][zslot] = 0.0f; dxt[1][zslot] = 0.0f; }

  const float* Ab  = A + (size_t)b * NN * NN + (size_t)r0 * NN;
  const float* dxb = dxw + (size_t)b * NN * ND;

  int buf = 0;

#if USE_ASYNC
  // ---- issue async staging of tile 0 straight into LDS (ASYNCcnt) ----
  {
#pragma unroll
    for (int q = 0; q < 8; ++q) {
      int lin = q * TPB + t;
      int srow = lin >> 4, scol = (lin & 15) << 2;
      __builtin_amdgcn_global_load_async_to_lds_b128(
          AS1(Ab + (size_t)srow * NN + scol), AS3(&At[0][srow * ATW + scol]), 0, 0);
    }
    if (t < TW * ND) dxt[0][t] = dxb[t];
  }
#else
  // ---- register prefetch of tile 0 ----
  float4 pa[8];
  float  pdx = 0.0f;
#pragma unroll
  for (int q = 0; q < 8; ++q) {
    int lin = q * TPB + t;
    int srow = lin >> 4, scol = (lin & 15) << 2;
    pa[q] = *(const float4*)(Ab + (size_t)srow * NN + scol);
  }
  if (t < TW * ND) pdx = dxb[t];
#endif

  v8f acc = {0.f, 0.f, 0.f, 0.f, 0.f, 0.f, 0.f, 0.f};

  for (int jt = 0; jt < NTILE; ++jt) {
#if USE_ASYNC
    if (jt + 1 < NTILE) {                 // issue next tile into other buffer
      const int j0 = (jt + 1) * TW;
#pragma unroll
      for (int q = 0; q < 8; ++q) {
        int lin = q * TPB + t;
        int srow = lin >> 4, scol = (lin & 15) << 2;
        __builtin_amdgcn_global_load_async_to_lds_b128(
            AS1(Ab + (size_t)srow * NN + j0 + scol),
            AS3(&At[buf ^ 1][srow * ATW + scol]), 0, 0);
      }
      if (t < TW * ND) dxt[buf ^ 1][t] = dxb[j0 * ND + t];
      __builtin_amdgcn_s_wait_asynccnt(8);   // current tile landed; next still in flight
    } else {
      __builtin_amdgcn_s_wait_asynccnt(0);
    }
    __syncthreads();                      // all waves see tile jt
#else
    __syncthreads();                      // previous buffer fully consumed
#pragma unroll
    for (int q = 0; q < 8; ++q) {
      int lin = q * TPB + t;
      int srow = lin >> 4, scol = (lin & 15) << 2;
      *(float4*)&At[buf][srow * ATW + scol] = pa[q];
    }
    if (t < TW * ND) dxt[buf][t] = pdx;
    __syncthreads();                      // tile visible
    if (jt + 1 < NTILE) {                 // prefetch next tile under compute
      const int j0 = (jt + 1) * TW;
#pragma unroll
      for (int q = 0; q < 8; ++q) {
        int lin = q * TPB + t;
        int srow = lin >> 4, scol = (lin & 15) << 2;
        pa[q] = *(const float4*)(Ab + (size_t)srow * NN + j0 + scol);
      }
      if (t < TW * ND) pdx = dxb[j0 * ND + t];
    }
#endif

    // ---- deg partial: 4 columns x 8 rows per thread, b128 LDS reads ----
    {
      float s0 = 0.f, s1 = 0.f, s2 = 0.f, s3 = 0.f;
#pragma unroll
      for (int r = 0; r < 8; ++r) {
        const float4 v = *(const float4*)&At[buf][(rg + r) * ATW + c4];
        s0 += v.x; s1 += v.y; s2 += v.z; s3 += v.w;
      }
      atomicAdd(&degl[jt * TW + c4 + 0], s0);
      atomicAdd(&degl[jt * TW + c4 + 1], s1);
      atomicAdd(&degl[jt * TW + c4 + 2], s2);
      atomicAdd(&degl[jt * TW + c4 + 3], s3);
    }

    // ---- 16 WMMA k-steps: 16 rows x 64 cols of A against dx tile ----
#pragma unroll
    for (int kb = 0; kb < TW; kb += 4) {
      float2 af = *(const float2*)&At[buf][aRowBase + kb];  // A[m][kb+kk], A[m][kb+kk+1]
      v2f av; av.x = af.x; av.y = af.y;
      int bi0 = nv ? ((kb + kk) * ND + mi) : zslot;         // B[k][n] = dx[j0+k][n], n<3
      int bi1 = nv ? ((kb + kk + 1) * ND + mi) : zslot;
      v2f bv; bv.x = dxt[buf][bi0]; bv.y = dxt[buf][bi1];
      acc = __builtin_amdgcn_wmma_f32_16x16x4_f32(
          /*neg_a=*/false, av, /*neg_b=*/false, bv,
          /*c_mod=*/(short)0, acc, /*reuse_a=*/false, /*reuse_b=*/false);
    }

#if USE_ASYNC
    __syncthreads();                      // buffer consumed; next issue may overwrite
#endif
    buf ^= 1;
  }

  // ---- write neigh: C layout lane<16 -> M=v, lane>=16 -> M=v+8, N=lane%16 ----
  if (nv) {
    int rowoff = r0 + wave * 16 + ((lane < 16) ? 0 : 8);
#pragma unroll
    for (int v = 0; v < 8; ++v)
      neighw[((size_t)b * NN + rowoff + v) * ND + mi] = acc[v];
  }

  __syncthreads();
  // ---- flush per-WG deg partials (plain coalesced stores, no global atomics) ----
  float* dp = degp + (size_t)blockIdx.x * NN;
  for (int q = 0; q < NN / 4 / TPB; ++q) {
    int i4 = q * TPB + t;
    *(float4*)&dp[i4 * 4] = *(const float4*)&degl[i4 * 4];
  }
}

__global__ void lap_finish(const float* __restrict__ dxw, const float* __restrict__ neighw,
                           const float* __restrict__ degp, float* __restrict__ out) {
  __shared__ float red[256];
  int gid = blockIdx.x * 256 + threadIdx.x;
  int b = gid >> 12;
  int i = gid & (NN - 1);
  float deg = 0.0f;
  for (int rbl = 0; rbl < WGPB; ++rbl)
    deg += degp[(size_t)(b * WGPB + rbl) * NN + i];
  float accv = 0.0f;
#pragma unroll
  for (int d = 0; d < ND; ++d) {
    size_t idx = ((size_t)b * NN + i) * ND + d;
    float e = dxw[idx] - neighw[idx] / deg;
    accv += e * e;
  }
  red[threadIdx.x] = accv;
  __syncthreads();
  for (int off = 128; off > 0; off >>= 1) {
    if (threadIdx.x < off) red[threadIdx.x] += red[threadIdx.x + off];
    __syncthreads();
  }
  if (threadIdx.x == 0) atomicAdd(out, red[0]);
}

extern "C" void kernel_launch(void* const* d_in, const int* in_sizes, int n_in,
                              void* d_out, int out_size, void* d_ws, size_t ws_size,
                              hipStream_t stream) {
  (void)in_sizes; (void)n_in; (void)out_size; (void)ws_size;
  const float* p1 = (const float*)d_in[0];
  const float* p2 = (const float*)d_in[1];
  const float* A  = (const float*)d_in[2];
  float* out = (float*)d_out;
  float* ws  = (float*)d_ws;

  float* dxw    = ws;                                   // 8*4096*3
  float* neighw = dxw + (size_t)NB * NN * ND;           // 8*4096*3
  float* degp   = neighw + (size_t)NB * NN * ND;        // 8*64*4096

  lap_prep<<<(NB * NN * ND + 255) / 256, 256, 0, stream>>>(p1, p2, dxw, out);
  lap_main<<<NB * WGPB, TPB, 0, stream>>>(A, dxw, neighw, degp);
  lap_finish<<<NB * NN / 256, 256, 0, stream>>>(dxw, neighw, degp, out);
}